// EdgeConv_53644141527057
// MI455X (gfx1250) — compile-verified
//
#include <hip/hip_runtime.h>
#include <hip/hip_bf16.h>

// ---------------------------------------------------------------------------
// EdgeConv (line-graph message passing) for MI455X / gfx1250, wave32.
//
// out = elu(edge_attr @ W_e.T + b_e)                        [kernel 1]
//     + segment_sum( elu([ea[e1] | x[a0] | gs[a0]] @ W_edge.T + b_edge), e0 )
//                                                            [kernel 2]
//
// Matrix math uses V_WMMA_F32_16X16X4_F32 (exact fp32, matches reference).
// Each wave owns a 16-row tile and four 16x16 f32 accumulators (64 out chans).
// K padded 106 -> 108 so k-steps of 4 divide evenly and LDS rows stay
// 16B-aligned (108 floats = 432 B = 27 * 16 B).
// ---------------------------------------------------------------------------

typedef __attribute__((ext_vector_type(2))) float v2f;
typedef __attribute__((ext_vector_type(4))) float v4f;
typedef __attribute__((ext_vector_type(8))) float v8f;

#define N_NODES   100000
#define N_EDGES   1000000
#define N_PAIRS   2000000
#define IN_CH     64
#define OUT_CH    64
#define NODE_DIM  37
#define STATE_DIM 5
#define FAN       106     // IN_CH + NODE_DIM + STATE_DIM
#define FAN_PAD   108     // padded to a multiple of 4 (WMMA K-step)

__device__ __forceinline__ float elu_f(float v) {
    return v > 0.0f ? v : (__expf(v) - 1.0f);
}

// ---------------------------------------------------------------------------
// Kernel 1: out[e] = elu(edge_attr[e] @ W_e.T + b_e)   (1M x 64 x 64 GEMM)
// 256 threads = 8 waves; each wave computes a 16x64 output tile.
// ---------------------------------------------------------------------------
__global__ __launch_bounds__(256) void base_gemm_kernel(
    const float* __restrict__ edge_attr,  // [N_EDGES][64]
    const float* __restrict__ W_e,        // [64][64]
    const float* __restrict__ b_e,        // [64]
    float* __restrict__ out)              // [N_EDGES][64]
{
    __shared__ __align__(16) float sW[64 * 64];   // 16 KB, B-operand source
    for (int i = threadIdx.x; i < 64 * 64; i += 256) sW[i] = W_e[i];
    __syncthreads();

    const int wave = threadIdx.x >> 5;
    const int lane = threadIdx.x & 31;
    const int tile = blockIdx.x * 8 + wave;        // 16 rows per tile
    if (tile >= N_EDGES / 16) return;              // uniform per wave

    const int row0  = tile * 16;
    const int m     = lane & 15;                   // A row / B col within tile
    const int khalf = (lane >> 4) * 2;             // lanes 16-31 hold K+2,K+3

    v8f acc[4] = {};
    const float* arow = edge_attr + (size_t)(row0 + m) * IN_CH;

    for (int kb = 0; kb < IN_CH; kb += 4) {
        v2f a = *(const v2f*)(arow + kb + khalf);          // global_load_b64
#pragma unroll
        for (int nt = 0; nt < 4; ++nt) {
            // B[k][n] = W_e[n][k]; lane supplies 2 consecutive k for col n
            v2f b = *(const v2f*)(&sW[(nt * 16 + m) * 64 + kb + khalf]);
            acc[nt] = __builtin_amdgcn_wmma_f32_16x16x4_f32(
                false, a, false, b, (short)0, acc[nt], false, false);
        }
    }

    // C/D layout: VGPR v, lanes 0-15 -> M=v, lanes 16-31 -> M=v+8; N=lane&15.
    const int mbase = row0 + 8 * (lane >> 4);
#pragma unroll
    for (int nt = 0; nt < 4; ++nt) {
        const int   n    = nt * 16 + m;
        const float bias = b_e[n];
#pragma unroll
        for (int v = 0; v < 8; ++v) {
            out[(size_t)(mbase + v) * OUT_CH + n] = elu_f(acc[nt][v] + bias);
        }
    }
}

// ---------------------------------------------------------------------------
// Kernel 2: per-pair gather -> [16x108] @ [108x64] WMMA -> ELU -> atomic
// scatter-add into out rows e_idx0.  128 threads = 4 waves per block,
// 16 pairs per wave.  LDS: W_edge zero-padded (27648 B) + 4 per-wave feature
// tiles (4 * 6912 B) = 55296 B per workgroup.
// ---------------------------------------------------------------------------
__global__ __launch_bounds__(128) void pair_gemm_scatter_kernel(
    const float* __restrict__ x,          // [N_NODES][37]
    const float* __restrict__ gs,         // [N_NODES][5]
    const float* __restrict__ edge_attr,  // [N_EDGES][64]
    const float* __restrict__ W_edge,     // [64][106]
    const float* __restrict__ b_edge,     // [64]
    const int*   __restrict__ atom0,      // atom_index[0][*]
    const int*   __restrict__ eidx0,      // e_idx[0][*]  (scatter target)
    const int*   __restrict__ eidx1,      // e_idx[1][*]  (gather source)
    float* __restrict__ out)              // [N_EDGES][64], accumulated
{
    __shared__ __align__(16) float sW[64 * FAN_PAD];        // 27648 B
    __shared__ __align__(16) float sF[4][16 * FAN_PAD];     // 4 x 6912 B

    // Stage W_edge transposed-access-friendly (row-major, K padded w/ zeros).
    for (int i = threadIdx.x; i < 64 * FAN_PAD; i += 128) {
        const int n = i / FAN_PAD, k = i % FAN_PAD;
        sW[i] = (k < FAN) ? W_edge[n * FAN + k] : 0.0f;
    }

    const int wave = threadIdx.x >> 5;
    const int lane = threadIdx.x & 31;
    const int tile = blockIdx.x * 4 + wave;   // exact: N_PAIRS/16 tiles total
    const int p0   = tile * 16;
    float* feat = &sF[wave][0];

    // Gather 16 feature rows: 2 lanes cooperate per row.
    {
        const int r    = lane >> 1;           // row 0..15
        const int half = lane & 1;
        const int pair = p0 + r;
        const int e1   = eidx1[pair];
        const int a0   = atom0[pair];
        float* frow = feat + r * FAN_PAD;

        // edge_attr part: 64 floats, 8 x b128 per half-lane
        const v4f* ev = (const v4f*)(edge_attr + (size_t)e1 * IN_CH);
        v4f* fv = (v4f*)frow;                  // 432 B row stride: 16B aligned
#pragma unroll
        for (int q = 0; q < 8; ++q) fv[half * 8 + q] = ev[half * 8 + q];

        // node part: x (37) then global_state (5) into cols 64..105
        const float* xr = x  + (size_t)a0 * NODE_DIM;
        const float* gr = gs + (size_t)a0 * STATE_DIM;
        for (int q = half * 21; q < half * 21 + 21; ++q)
            frow[IN_CH + q] = (q < NODE_DIM) ? xr[q] : gr[q - NODE_DIM];

        if (half == 0) { frow[FAN] = 0.0f; frow[FAN + 1] = 0.0f; }  // pad
    }
    __syncthreads();   // sW visible to all waves; feat writes ordered vs reads

    const int m     = lane & 15;
    const int khalf = (lane >> 4) * 2;

    v8f acc[4] = {};
    for (int kb = 0; kb < FAN_PAD; kb += 4) {               // 27 k-steps
        v2f a = *(const v2f*)(feat + m * FAN_PAD + kb + khalf);  // ds_load_b64
#pragma unroll
        for (int nt = 0; nt < 4; ++nt) {
            v2f b = *(const v2f*)(&sW[(nt * 16 + m) * FAN_PAD + kb + khalf]);
            acc[nt] = __builtin_amdgcn_wmma_f32_16x16x4_f32(
                false, a, false, b, (short)0, acc[nt], false, false);
        }
    }

    // Scatter: row M = v + 8*(lane>=16) of this tile goes to edge e_idx0[..].
    const int mbase = p0 + 8 * (lane >> 4);
    int dest[8];
#pragma unroll
    for (int v = 0; v < 8; ++v) dest[v] = eidx0[mbase + v];

#pragma unroll
    for (int nt = 0; nt < 4; ++nt) {
        const int   n    = nt * 16 + m;
        const float bias = b_edge[n];
#pragma unroll
        for (int v = 0; v < 8; ++v) {
            const float h = elu_f(acc[nt][v] + bias);
            atomicAdd(&out[(size_t)dest[v] * OUT_CH + n], h);  // f32 L2 atomic
        }
    }
}

// ---------------------------------------------------------------------------
extern "C" void kernel_launch(void* const* d_in, const int* in_sizes, int n_in,
                              void* d_out, int out_size, void* d_ws, size_t ws_size,
                              hipStream_t stream) {
    const float* x      = (const float*)d_in[0];
    const float* gstate = (const float*)d_in[1];
    const float* eattr  = (const float*)d_in[2];
    const float* W_edge = (const float*)d_in[3];
    const float* b_edge = (const float*)d_in[4];
    const float* W_e    = (const float*)d_in[5];
    const float* b_e    = (const float*)d_in[6];
    const int*   atom   = (const int*)d_in[7];   // [2][N_PAIRS]
    const int*   eidx   = (const int*)d_in[8];   // [2][N_PAIRS]
    float* out = (float*)d_out;

    // Kernel 1 initializes every output element (elu(ea @ W_e.T + b_e)).
    const int tiles1  = N_EDGES / 16;            // 62500
    const int blocks1 = (tiles1 + 7) / 8;        // 8 waves/block
    base_gemm_kernel<<<blocks1, 256, 0, stream>>>(eattr, W_e, b_e, out);

    // Kernel 2 accumulates the scattered pair messages (stream-ordered).
    const int blocks2 = (N_PAIRS / 16) / 4;      // 31250, exact
    pair_gemm_scatter_kernel<<<blocks2, 128, 0, stream>>>(
        x, gstate, eattr, W_edge, b_edge,
        atom /*row 0*/, eidx /*row 0*/, eidx + N_PAIRS /*row 1*/, out);
}